// Decoder_66443144069661
// MI455X (gfx1250) — compile-verified
//
#include <hip/hip_runtime.h>
#include <math.h>

typedef __attribute__((ext_vector_type(2))) float v2f;
typedef __attribute__((ext_vector_type(8))) float v8f;
typedef __attribute__((ext_vector_type(4))) unsigned int v4u;
typedef __attribute__((ext_vector_type(4))) int v4i;
typedef __attribute__((ext_vector_type(8))) int v8i;

#define S 256
#define BATCH 16
#define LDIM 36
#define NPTS 400000
#define NATOMS 50000
#define NBOND 50000
#define NANG 50000
#define IMG_ELEMS (BATCH * S * S)   // 1048576

#if defined(__HIP_DEVICE_COMPILE__) && __has_builtin(__builtin_amdgcn_tensor_load_to_lds)
#define USE_TDM 1
#else
#define USE_TDM 0
#endif

// ---------------------------------------------------------------------------
// Setup: per-batch rotation matrix rows + shifts into ws (12 floats / batch)
// ---------------------------------------------------------------------------
__global__ void prep_rs(const float* __restrict__ euler,
                        const float* __restrict__ shifts,
                        float* __restrict__ RS) {
  int b = threadIdx.x;
  if (b >= BATCH) return;
  float a = euler[b * 3 + 0], bb = euler[b * 3 + 1], g = euler[b * 3 + 2];
  float sa, ca, sb, cb, sg, cg;
  sincosf(a, &sa, &ca);
  sincosf(bb, &sb, &cb);
  sincosf(g, &sg, &cg);
  float* r = RS + b * 12;
  r[0] = cg * cb * ca - sg * sa;
  r[1] = cg * cb * sa + sg * ca;
  r[2] = -cg * sb;
  r[3] = -sg * cb * ca - cg * sa;
  r[4] = -sg * cb * sa + cg * ca;
  r[5] = sg * sb;
  r[6] = sb * ca;
  r[7] = sb * sa;
  r[8] = cb;
  r[9] = shifts[b * 2 + 0];
  r[10] = shifts[b * 2 + 1];
  r[11] = 0.0f;
}

__global__ void zero_f32(float* __restrict__ p, int n) {
  int i = blockIdx.x * blockDim.x + threadIdx.x;
  if (i < n) p[i] = 0.0f;
}

// ---------------------------------------------------------------------------
// Fused deformation GEMM (WMMA f32 16x16x4) + rotate + shift + bilinear
// scatter.  One wave = 16 points x 16 batches.  A = z (M=batch), B = Z_vol
// tile (N=point), K = L = 36 = 9 steps of 4.
// ---------------------------------------------------------------------------
__global__ void deform_rotate_scatter(const float* __restrict__ z_x,
                                      const float* __restrict__ z_y,
                                      const float* __restrict__ z_z,
                                      const float* __restrict__ Zvol,
                                      const float* __restrict__ coords,
                                      const float* __restrict__ values,
                                      const float* __restrict__ RS,
                                      float* __restrict__ img, int n) {
  int wave = threadIdx.x >> 5;
  int lane = threadIdx.x & 31;
  int tile = blockIdx.x * (blockDim.x >> 5) + wave;
  int nTiles = n >> 4;
  if (tile >= nTiles) return;

  int lh = lane >> 4;   // 0 or 1 (K-half)
  int ln = lane & 15;   // M row for A, N col for B
  int p = (tile << 4) + ln;

  const float2* zx2 = (const float2*)z_x;   // row stride 18 float2
  const float2* zy2 = (const float2*)z_y;
  const float2* zz2 = (const float2*)z_z;
  const float2* zv2 = (const float2*)Zvol;

  __builtin_prefetch(&Zvol[p * LDIM], 0, 3);

  v8f ax = {}, ay = {}, az = {};
#pragma unroll
  for (int k = 0; k < 9; ++k) {
    int kb2 = k * 2 + lh;  // float2 index within a 36-float row
    float2 t;
    v2f A0, A1, A2, Bv;
    t = zx2[ln * 18 + kb2]; A0.x = t.x; A0.y = t.y;
    t = zy2[ln * 18 + kb2]; A1.x = t.x; A1.y = t.y;
    t = zz2[ln * 18 + kb2]; A2.x = t.x; A2.y = t.y;
    t = zv2[p * 18 + kb2];  Bv.x = t.x; Bv.y = t.y;
    ax = __builtin_amdgcn_wmma_f32_16x16x4_f32(false, A0, false, Bv, (short)0, ax, false, false);
    ay = __builtin_amdgcn_wmma_f32_16x16x4_f32(false, A1, false, Bv, (short)0, ay, false, false);
    az = __builtin_amdgcn_wmma_f32_16x16x4_f32(false, A2, false, Bv, (short)0, az, false, false);
  }

  float c0 = coords[3 * p + 0];
  float c1 = coords[3 * p + 1];
  float c2 = coords[3 * p + 2];
  float val = values[p];

#pragma unroll
  for (int r = 0; r < 8; ++r) {
    int b = r + (lh << 3);
    const float* Rm = RS + b * 12;
    float cx = c0 + ax[r];
    float cy = c1 + ay[r];
    float cz = c2 + az[r];
    float rx = Rm[0] * cx + Rm[1] * cy + Rm[2] * cz + Rm[9];
    float ry = Rm[3] * cx + Rm[4] * cy + Rm[5] * cz + Rm[10];
    float fx = rx + 128.0f, fy = ry + 128.0f;
    float x0 = floorf(fx), y0 = floorf(fy);
    float wx = fx - x0, wy = fy - y0;
    int xi = (int)x0, yi = (int)y0;
    float* ib = img + b * (S * S);
    if ((unsigned)xi < (unsigned)S && (unsigned)yi < (unsigned)S)
      atomicAdd(ib + yi * S + xi, (1.0f - wx) * (1.0f - wy) * val);
    if ((unsigned)(xi + 1) < (unsigned)S && (unsigned)yi < (unsigned)S)
      atomicAdd(ib + yi * S + xi + 1, wx * (1.0f - wy) * val);
    if ((unsigned)xi < (unsigned)S && (unsigned)(yi + 1) < (unsigned)S)
      atomicAdd(ib + (yi + 1) * S + xi, (1.0f - wx) * wy * val);
    if ((unsigned)(xi + 1) < (unsigned)S && (unsigned)(yi + 1) < (unsigned)S)
      atomicAdd(ib + (yi + 1) * S + xi + 1, wx * wy * val);
  }
}

// ---------------------------------------------------------------------------
// Atom positions: a[b, p, :] = atom_coords[p] + (z @ Z_atoms.T)[b, p]
// ---------------------------------------------------------------------------
__global__ void deform_atoms(const float* __restrict__ z_x,
                             const float* __restrict__ z_y,
                             const float* __restrict__ z_z,
                             const float* __restrict__ Zat,
                             const float* __restrict__ atom_coords,
                             float* __restrict__ aPos, int n) {
  int wave = threadIdx.x >> 5;
  int lane = threadIdx.x & 31;
  int tile = blockIdx.x * (blockDim.x >> 5) + wave;
  int nTiles = n >> 4;
  if (tile >= nTiles) return;

  int lh = lane >> 4;
  int ln = lane & 15;
  int p = (tile << 4) + ln;

  const float2* zx2 = (const float2*)z_x;
  const float2* zy2 = (const float2*)z_y;
  const float2* zz2 = (const float2*)z_z;
  const float2* za2 = (const float2*)Zat;

  __builtin_prefetch(&Zat[p * LDIM], 0, 3);

  v8f ax = {}, ay = {}, az = {};
#pragma unroll
  for (int k = 0; k < 9; ++k) {
    int kb2 = k * 2 + lh;
    float2 t;
    v2f A0, A1, A2, Bv;
    t = zx2[ln * 18 + kb2]; A0.x = t.x; A0.y = t.y;
    t = zy2[ln * 18 + kb2]; A1.x = t.x; A1.y = t.y;
    t = zz2[ln * 18 + kb2]; A2.x = t.x; A2.y = t.y;
    t = za2[p * 18 + kb2];  Bv.x = t.x; Bv.y = t.y;
    ax = __builtin_amdgcn_wmma_f32_16x16x4_f32(false, A0, false, Bv, (short)0, ax, false, false);
    ay = __builtin_amdgcn_wmma_f32_16x16x4_f32(false, A1, false, Bv, (short)0, ay, false, false);
    az = __builtin_amdgcn_wmma_f32_16x16x4_f32(false, A2, false, Bv, (short)0, az, false, false);
  }

  float c0 = atom_coords[3 * p + 0];
  float c1 = atom_coords[3 * p + 1];
  float c2 = atom_coords[3 * p + 2];

#pragma unroll
  for (int r = 0; r < 8; ++r) {
    int b = r + (lh << 3);
    float* dst = aPos + ((size_t)b * n + p) * 3;
    dst[0] = c0 + ax[r];
    dst[1] = c1 + ay[r];
    dst[2] = c2 + az[r];
  }
}

// ---------------------------------------------------------------------------
// Bond lengths / bend angles from gathered atom positions
// ---------------------------------------------------------------------------
__global__ void bond_kernel(const float* __restrict__ aPos,
                            const int* __restrict__ bidx,
                            float* __restrict__ out) {
  int idx = blockIdx.x * blockDim.x + threadIdx.x;
  if (idx >= BATCH * NBOND) return;
  int j = idx % NBOND;
  int b = idx / NBOND;
  int i0 = bidx[2 * j + 0];
  int i1 = bidx[2 * j + 1];
  const float* base = aPos + (size_t)b * NATOMS * 3;
  float dx = base[i0 * 3 + 0] - base[i1 * 3 + 0];
  float dy = base[i0 * 3 + 1] - base[i1 * 3 + 1];
  float dz = base[i0 * 3 + 2] - base[i1 * 3 + 2];
  out[idx] = sqrtf(dx * dx + dy * dy + dz * dz + 1e-12f);
}

__global__ void angle_kernel(const float* __restrict__ aPos,
                             const int* __restrict__ aidx,
                             float* __restrict__ out) {
  int idx = blockIdx.x * blockDim.x + threadIdx.x;
  if (idx >= BATCH * NANG) return;
  int j = idx % NANG;
  int b = idx / NANG;
  int i0 = aidx[3 * j + 0];
  int i1 = aidx[3 * j + 1];
  int i2 = aidx[3 * j + 2];
  const float* base = aPos + (size_t)b * NATOMS * 3;
  float ux = base[i0 * 3 + 0] - base[i1 * 3 + 0];
  float uy = base[i0 * 3 + 1] - base[i1 * 3 + 1];
  float uz = base[i0 * 3 + 2] - base[i1 * 3 + 2];
  float vx = base[i2 * 3 + 0] - base[i1 * 3 + 0];
  float vy = base[i2 * 3 + 1] - base[i1 * 3 + 1];
  float vz = base[i2 * 3 + 2] - base[i1 * 3 + 2];
  float uu = ux * ux + uy * uy + uz * uz;
  float vv = vx * vx + vy * vy + vz * vz;
  float uv = ux * vx + uy * vy + uz * vz;
  float cosang = uv / (sqrtf(uu * vv) + 1e-6f);
  cosang = fminf(fmaxf(cosang, -0.999f), 0.999f);
  out[idx] = acosf(cosang);
}

// ---------------------------------------------------------------------------
// 5x5 Gaussian conv (SAME, zero pad), separable weights hardcoded (sigma=1)
// ---------------------------------------------------------------------------
__global__ void conv5(const float* __restrict__ in, float* __restrict__ out) {
  int idx = blockIdx.x * blockDim.x + threadIdx.x;
  if (idx >= IMG_ELEMS) return;
  int x = idx & (S - 1);
  int y = (idx >> 8) & (S - 1);
  int b = idx >> 16;
  const float w1d[5] = {0.05448868f, 0.24420134f, 0.40261996f, 0.24420134f, 0.05448868f};
  const float* ib = in + b * (S * S);
  float acc = 0.0f;
#pragma unroll
  for (int dy = 0; dy < 5; ++dy) {
    int yy = y + dy - 2;
    if ((unsigned)yy >= (unsigned)S) continue;
#pragma unroll
    for (int dx = 0; dx < 5; ++dx) {
      int xx = x + dx - 2;
      if ((unsigned)xx >= (unsigned)S) continue;
      acc += w1d[dy] * w1d[dx] * ib[yy * S + xx];
    }
  }
  out[idx] = acc;
}

// ---------------------------------------------------------------------------
// 256-point radix-2 DIT FFT in LDS.  128 threads = 1 butterfly each / stage.
// Input must be loaded in bit-reversed order.  sign=-1 fwd, +1 inv (unscaled).
// ---------------------------------------------------------------------------
__device__ __forceinline__ void fft256_stages(float2* s, int tid, float sign) {
#pragma unroll
  for (int len = 2; len <= 256; len <<= 1) {
    int hl = len >> 1;
    int k = tid & (hl - 1);
    int i0 = (tid / hl) * len + k;
    float ang = sign * 6.28318530717958647692f * (float)k / (float)len;
    float sn, c;
    __sincosf(ang, &sn, &c);
    float2 u = s[i0];
    float2 v = s[i0 + hl];
    float wr = v.x * c - v.y * sn;
    float wi = v.x * sn + v.y * c;
    s[i0] = make_float2(u.x + wr, u.y + wi);
    s[i0 + hl] = make_float2(u.x - wr, u.y - wi);
    __syncthreads();
  }
}

__global__ void fft_rows_fwd(const float* __restrict__ in, float2* __restrict__ out) {
  __shared__ float2 s[S];
  int line = blockIdx.x;
  const float* src = in + (size_t)line * S;
  int tid = threadIdx.x;
  for (int i = tid; i < S; i += 128) {
    int r = __brev((unsigned)i) >> 24;
    s[i] = make_float2(src[r], 0.0f);
  }
  __syncthreads();
  fft256_stages(s, tid, -1.0f);
  float2* dst = out + (size_t)line * S;
  for (int i = tid; i < S; i += 128) dst[i] = s[i];
}

// Column pass: TDM DMA of the strided column tile into LDS (tile 2x256 f32,
// line stride 512 f32), then fwd FFT -> *ctf -> inv FFT, store back.
__global__ void fft_cols_ctf(float2* __restrict__ data, const float* __restrict__ ctf) {
  __shared__ float2 s[S];
  int col = blockIdx.x & (S - 1);
  int b = blockIdx.x >> 8;
  float2* base = data + (size_t)b * S * S + col;
  int tid = threadIdx.x;

#if USE_TDM
  if (tid < 32) {  // one wave issues the tensor DMA (block-uniform descriptor)
    unsigned long long ga = (unsigned long long)(uintptr_t)base;
    unsigned lds = (unsigned)(uintptr_t)&s[0];
    v4u g0;
    g0.x = 1u;                                   // count=1, user mode
    g0.y = lds;                                  // lds_addr
    g0.z = (unsigned)(ga & 0xffffffffu);         // global_addr[31:0]
    g0.w = (unsigned)((ga >> 32) & 0x01ffffffu)  // global_addr[56:32]
         | (2u << 30);                           // type=2 (image)
    v8i g1;
    g1[0] = (int)(2u << 16);    // data_size=2 (4B), mask=0, no flags
    g1[1] = (int)(2u << 16);    // tensor_dim0 = 2 (bits 63:48)
    g1[2] = (int)(256u << 16);  // tensor_dim1 = 256 (bits 95:80)
    g1[3] = (int)(2u << 16);    // tile_dim0 = 2 (bits 127:112)
    g1[4] = 256;                // tile_dim1 = 256 (bits 143:128)
    g1[5] = 512;                // tensor_dim0_stride = 512 (bits 191:160)
    g1[6] = 0;
    g1[7] = 0;
    v4i gz = {0, 0, 0, 0};      // groups 2/3 unused (2D tile)
#if defined(__clang_major__) && __clang_major__ >= 23
    v8i gz8 = {0, 0, 0, 0, 0, 0, 0, 0};
    __builtin_amdgcn_tensor_load_to_lds(g0, g1, gz, gz, gz8, 0);
#else
    __builtin_amdgcn_tensor_load_to_lds(g0, g1, gz, gz, 0);
#endif
    __builtin_amdgcn_s_wait_tensorcnt(0);
  }
  __syncthreads();
  // bit-reverse permute (natural -> DIT order) in place
  {
    float2 e0 = s[tid];
    float2 e1 = s[tid + 128];
    __syncthreads();
    s[__brev((unsigned)tid) >> 24] = e0;
    s[__brev((unsigned)(tid + 128)) >> 24] = e1;
    __syncthreads();
  }
#else
  for (int i = tid; i < S; i += 128) {
    int r = __brev((unsigned)i) >> 24;
    s[i] = base[(size_t)r * S];
  }
  __syncthreads();
#endif

  fft256_stages(s, tid, -1.0f);       // forward column FFT (natural order now)
  const float* cb = ctf + (size_t)b * S * S + col;
  for (int i = tid; i < S; i += 128) {
    float t = cb[(size_t)i * S];
    s[i].x *= t;
    s[i].y *= t;
  }
  __syncthreads();
  // bit-reverse permute in place for the inverse pass
  float2 e0 = s[tid];
  float2 e1 = s[tid + 128];
  __syncthreads();
  s[__brev((unsigned)tid) >> 24] = e0;
  s[__brev((unsigned)(tid + 128)) >> 24] = e1;
  __syncthreads();
  fft256_stages(s, tid, 1.0f);        // inverse column FFT (unscaled)
  for (int i = tid; i < S; i += 128) base[(size_t)i * S] = s[i];
}

__global__ void ifft_rows_real(const float2* __restrict__ data, float* __restrict__ out) {
  __shared__ float2 s[S];
  int line = blockIdx.x;
  const float2* src = data + (size_t)line * S;
  int tid = threadIdx.x;
  for (int i = tid; i < S; i += 128) {
    int r = __brev((unsigned)i) >> 24;
    s[i] = src[r];
  }
  __syncthreads();
  fft256_stages(s, tid, 1.0f);        // inverse row FFT (unscaled)
  float* dst = out + (size_t)line * S;
  const float sc = 1.0f / 65536.0f;   // 1/(S*S) total ifft2 scaling
  for (int i = tid; i < S; i += 128) dst[i] = s[i].x * sc;
}

// ---------------------------------------------------------------------------
// Launcher
// ---------------------------------------------------------------------------
extern "C" void kernel_launch(void* const* d_in, const int* in_sizes, int n_in,
                              void* d_out, int out_size, void* d_ws, size_t ws_size,
                              hipStream_t stream) {
  const float* z_x         = (const float*)d_in[0];
  const float* z_y         = (const float*)d_in[1];
  const float* z_z         = (const float*)d_in[2];
  const float* delta_euler = (const float*)d_in[3];
  const float* delta_shift = (const float*)d_in[4];
  const float* Z_vol       = (const float*)d_in[5];
  const float* Z_atoms     = (const float*)d_in[6];
  const float* coords      = (const float*)d_in[7];
  const float* values      = (const float*)d_in[8];
  const float* atom_coords = (const float*)d_in[9];
  const float* ctf         = (const float*)d_in[10];
  const int*   bond_idx    = (const int*)d_in[11];
  const int*   angle_idx   = (const int*)d_in[12];

  // workspace layout (floats)
  float*  ws    = (float*)d_ws;
  float*  RS    = ws;                         // 256 floats (192 used)
  float*  img   = ws + 256;                   // 1048576
  float*  cimg  = img + IMG_ELEMS;            // 1048576
  float2* freq  = (float2*)(cimg + IMG_ELEMS);// 1048576 float2
  float*  aPos  = (float*)(freq + IMG_ELEMS); // 2400000

  float* outImg   = (float*)d_out;
  float* outBond  = outImg + IMG_ELEMS;
  float* outAngle = outBond + BATCH * NBOND;

  prep_rs<<<1, 16, 0, stream>>>(delta_euler, delta_shift, RS);
  zero_f32<<<IMG_ELEMS / 256, 256, 0, stream>>>(img, IMG_ELEMS);

  // 8 waves/block, 16 points/wave: 400000/16/8 = 3125 blocks
  deform_rotate_scatter<<<3125, 256, 0, stream>>>(z_x, z_y, z_z, Z_vol, coords,
                                                  values, RS, img, NPTS);
  // 50000/16 = 3125 tiles -> ceil(3125/8) = 391 blocks
  deform_atoms<<<391, 256, 0, stream>>>(z_x, z_y, z_z, Z_atoms, atom_coords,
                                        aPos, NATOMS);

  bond_kernel<<<(BATCH * NBOND + 255) / 256, 256, 0, stream>>>(aPos, bond_idx, outBond);
  angle_kernel<<<(BATCH * NANG + 255) / 256, 256, 0, stream>>>(aPos, angle_idx, outAngle);

  conv5<<<IMG_ELEMS / 256, 256, 0, stream>>>(img, cimg);

  fft_rows_fwd<<<BATCH * S, 128, 0, stream>>>(cimg, freq);
  fft_cols_ctf<<<BATCH * S, 128, 0, stream>>>(freq, ctf);
  ifft_rows_real<<<BATCH * S, 128, 0, stream>>>(freq, outImg);
}